// Aggregator_53455162966709
// MI455X (gfx1250) — compile-verified
//
#include <hip/hip_runtime.h>

#define DDIM 128
#define NREL 24

// ---------------------------------------------------------------------------
// gfx1250 async global->LDS staging (guarded: falls back to plain copy if the
// toolchain doesn't declare the builtin). Signature probe-confirmed in R1/R2:
//   void __builtin_amdgcn_global_load_async_to_lds_b128(
//       int __attribute__((vector_size(16))) AS1* gsrc,
//       int __attribute__((vector_size(16))) AS3* ldst,
//       imm int offset, imm int cpol);
// ---------------------------------------------------------------------------
#if defined(__AMDGCN__) && defined(__gfx1250__) && \
    __has_builtin(__builtin_amdgcn_global_load_async_to_lds_b128)
#define USE_ASYNC_STAGE 1
typedef int async_v4i __attribute__((vector_size(16)));
typedef __attribute__((address_space(1))) async_v4i* async_gptr_t;
typedef __attribute__((address_space(3))) async_v4i* async_lptr_t;
#endif

__device__ __forceinline__ void wait_asynccnt0() {
#if defined(__AMDGCN__) && defined(__gfx1250__)
#if __has_builtin(__builtin_amdgcn_s_wait_asynccnt)
  __builtin_amdgcn_s_wait_asynccnt(0);
#else
  asm volatile("s_wait_asynccnt 0" ::: "memory");
#endif
#endif
}

// ---------------------------------------------------------------------------
// Hot kernel (placed first so the disasm snippet shows it).
// Wave-per-edge-chunk: each wave owns 32 contiguous edges. Lane l loads the
// indices of edge base+l (coalesced 256B bursts), then the wave iterates the
// 32 edges, broadcasting head/tail/rel via uniform-index __shfl (v_readlane).
// Per edge: 512B coalesced row gather (b128/lane), multiply by the LDS-staged
// relation row (ds_load_b128), scatter via non-returning agent-scope
// global_atomic_add_f32 into the L2-resident output (51.6MB << 192MB L2).
// ---------------------------------------------------------------------------
__global__ __launch_bounds__(256) void scatter_kernel(
    const float* __restrict__ ego,        // [N, 128]
    const long long* __restrict__ eidx,   // [2, E] (row 0 = head, row 1 = tail)
    const long long* __restrict__ etype,  // [E]
    const float* __restrict__ relw,       // [24, 128]
    float* __restrict__ agg,              // [N, 128] (pre-zeroed)
    float* __restrict__ cnt,              // [N]      (pre-zeroed)
    int n_edges) {
  __shared__ float s_rel[NREL * DDIM];  // 12 KB of the 320 KB/WGP LDS

#ifdef USE_ASYNC_STAGE
  // 768 x b128 async DMA transfers into LDS, then drain ASYNCcnt.
  for (int i = threadIdx.x * 4; i < NREL * DDIM; i += blockDim.x * 4) {
    __builtin_amdgcn_global_load_async_to_lds_b128(
        (async_gptr_t)(relw + i), (async_lptr_t)(s_rel + i), 0, 0);
  }
  wait_asynccnt0();
#else
  for (int i = threadIdx.x; i < NREL * DDIM; i += blockDim.x)
    s_rel[i] = relw[i];
#endif
  __syncthreads();

  const int lane = threadIdx.x & 31;
  const int wave = threadIdx.x >> 5;
  const int waves_per_block = blockDim.x >> 5;
  const int chunk_stride = gridDim.x * waves_per_block * 32;

  for (int base = (blockIdx.x * waves_per_block + wave) * 32; base < n_edges;
       base += chunk_stride) {
    // Coalesced per-lane index loads for this wave's 32 edges.
    const int my_e = base + lane;
    int my_head = 0, my_tail = 0, my_rel = 0;
    if (my_e < n_edges) {
      my_head = (int)eidx[my_e];            // row 0
      my_tail = (int)eidx[n_edges + my_e];  // row 1
      my_rel = (int)etype[my_e];
    }
    const int mcount = (n_edges - base < 32) ? (n_edges - base) : 32;

    for (int j = 0; j < mcount; ++j) {
      // Uniform-index broadcast -> v_readlane_b32, no memory traffic.
      const int head = __shfl(my_head, j, 32);
      const int tail = __shfl(my_tail, j, 32);
      const int rel = __shfl(my_rel, j, 32);

      // 512-byte coalesced row gather: one b128 per lane.
      const float4 t =
          ((const float4*)(ego + (long long)tail * DDIM))[lane];
      const float4 w = ((const float4*)(s_rel + rel * DDIM))[lane];

      float* dst = agg + (long long)head * DDIM + lane * 4;
      __hip_atomic_fetch_add(dst + 0, t.x * w.x, __ATOMIC_RELAXED,
                             __HIP_MEMORY_SCOPE_AGENT);
      __hip_atomic_fetch_add(dst + 1, t.y * w.y, __ATOMIC_RELAXED,
                             __HIP_MEMORY_SCOPE_AGENT);
      __hip_atomic_fetch_add(dst + 2, t.z * w.z, __ATOMIC_RELAXED,
                             __HIP_MEMORY_SCOPE_AGENT);
      __hip_atomic_fetch_add(dst + 3, t.w * w.w, __ATOMIC_RELAXED,
                             __HIP_MEMORY_SCOPE_AGENT);
      if (lane == 0) {
        __hip_atomic_fetch_add(cnt + head, 1.0f, __ATOMIC_RELAXED,
                               __HIP_MEMORY_SCOPE_AGENT);
      }
    }
  }
}

// ---------------------------------------------------------------------------
// Zero the accumulator (d_out) and the per-entity counts (d_ws).
// ---------------------------------------------------------------------------
__global__ void zero_kernel(float4* __restrict__ out4, long long n4,
                            float* __restrict__ cnt, int n_cnt) {
  long long i = (long long)blockIdx.x * blockDim.x + threadIdx.x;
  long long stride = (long long)gridDim.x * blockDim.x;
  const float4 z = make_float4(0.f, 0.f, 0.f, 0.f);
  for (long long k = i; k < n4; k += stride) out4[k] = z;
  for (long long k = i; k < (long long)n_cnt; k += stride) cnt[k] = 0.0f;
}

// ---------------------------------------------------------------------------
// Wave-per-row mean: out[row] *= 1 / max(cnt[row], 1).
// ---------------------------------------------------------------------------
__global__ __launch_bounds__(256) void finalize_kernel(
    float* __restrict__ out, const float* __restrict__ cnt, int n_entities) {
  const int lane = threadIdx.x & 31;
  const int wave = threadIdx.x >> 5;
  const int row = blockIdx.x * (blockDim.x >> 5) + wave;
  if (row >= n_entities) return;
  const float inv = 1.0f / fmaxf(cnt[row], 1.0f);
  float4* p = (float4*)(out + (long long)row * DDIM);
  float4 v = p[lane];
  v.x *= inv;
  v.y *= inv;
  v.z *= inv;
  v.w *= inv;
  p[lane] = v;
}

// ---------------------------------------------------------------------------
// Host launcher. Inputs (setup_inputs order):
//   d_in[0] = ego_embed       float32 [N, 128]
//   d_in[1] = edge_index      int64   [2, E]
//   d_in[2] = edge_type       int64   [E]
//   d_in[3] = relation_weight float32 [24, 128]
// d_out = float32 [N, 128]; d_ws holds the per-entity counts (N floats).
// ---------------------------------------------------------------------------
extern "C" void kernel_launch(void* const* d_in, const int* in_sizes, int n_in,
                              void* d_out, int out_size, void* d_ws,
                              size_t ws_size, hipStream_t stream) {
  const float* ego = (const float*)d_in[0];
  const long long* eidx = (const long long*)d_in[1];
  const long long* etype = (const long long*)d_in[2];
  const float* relw = (const float*)d_in[3];
  float* out = (float*)d_out;
  float* cnt = (float*)d_ws;

  const int n_edges = in_sizes[2];            // E
  const int n_entities = in_sizes[0] / DDIM;  // N

  // 1) zero accumulator + counts
  const long long n4 = (long long)n_entities * (DDIM / 4);
  int zgrid = (int)((n4 + 255) / 256);
  zero_kernel<<<zgrid, 256, 0, stream>>>((float4*)out, n4, cnt, n_entities);

  // 2) scatter: 8 waves/block, 32 edges per wave-chunk; cap the grid so the
  //    12KB LDS staging amortizes over >=256 edges per block.
  const int wpb = 256 / 32;
  const int edges_per_block = wpb * 32;
  int sgrid = (n_edges + edges_per_block - 1) / edges_per_block;
  if (sgrid > 4096) sgrid = 4096;
  scatter_kernel<<<sgrid, 256, 0, stream>>>(ego, eidx, etype, relw, out, cnt,
                                            n_edges);

  // 3) mean
  int fgrid = (n_entities + wpb - 1) / wpb;
  finalize_kernel<<<fgrid, 256, 0, stream>>>(out, cnt, n_entities);
}